// MeanSquaredError2___56547539419667
// MI455X (gfx1250) — compile-verified
//
#include <hip/hip_runtime.h>

#define NJ 14
#define COL 14
#define CCH 18          // 14 joint + 4 group channels
#define RADIUS 4
#define WAVES_PER_BLOCK 8

typedef __attribute__((ext_vector_type(2))) float v2f;
typedef __attribute__((ext_vector_type(8))) float v8f;

// ---------------- workspace layout ----------------
// d_ws + 0  : double  loss accumulator
// d_ws + 8  : int     v.sum()

__global__ void init_ws_kernel(double* acc, int* vsum) {
    if (threadIdx.x == 0 && blockIdx.x == 0) {
        *acc = 0.0;
        *vsum = 0;
    }
}

__global__ void vsum_kernel(const int* __restrict__ v, int n, int* __restrict__ vsum) {
    int i = blockIdx.x * blockDim.x + threadIdx.x;
    int stride = gridDim.x * blockDim.x;
    int s = 0;
    for (; i < n; i += stride) s += v[i];
    #pragma unroll
    for (int off = 16; off; off >>= 1) s += __shfl_xor(s, off, 32);
    if ((threadIdx.x & 31) == 0 && s != 0) atomicAdd(vsum, s);
}

__global__ __launch_bounds__(32 * WAVES_PER_BLOCK)
void heatmap_mse_kernel(const float* __restrict__ h,
                        const float* __restrict__ t,
                        const int*   __restrict__ v,
                        double* __restrict__ acc,
                        int batch) {
    __shared__ float Gs[16][17];                        // padded 14x14 Gaussian matrix (reflect folded in)
    __shared__ float Tile[WAVES_PER_BLOCK][16][17];     // per-wave D->A relayout tile

    const int tid = threadIdx.x;

    // ---- build G (one entry per thread; rows/cols 14,15 are zero padding) ----
    {
        float k[2 * RADIUS + 1];
        float ksum = 0.f;
        #pragma unroll
        for (int i = 0; i <= 2 * RADIUS; ++i) {
            float x = (float)(i - RADIUS);
            k[i] = expf(-0.5f * x * x);
            ksum += k[i];
        }
        int y  = tid >> 4;
        int yy = tid & 15;
        float g = 0.f;
        if (y < COL && yy < COL) {
            #pragma unroll
            for (int d = -RADIUS; d <= RADIUS; ++d) {
                int p = y + d;
                if (p < 0)    p = -p - 1;          // 'symmetric' reflection
                if (p >= COL) p = 2 * COL - 1 - p;
                if (p == yy)  g += k[d + RADIUS];
            }
            g /= ksum;
        }
        Gs[y][yy] = g;
    }
    __syncthreads();

    const int wave = tid >> 5;
    const int lane = tid & 31;
    const long long gidx  = (long long)blockIdx.x * WAVES_PER_BLOCK + wave;
    const long long total = (long long)batch * CCH;
    if (gidx >= total) return;                    // wave-uniform exit

    const int b = (int)(gidx / CCH);
    const int c = (int)(gidx % CCH);

    // ---- gather up to 3 joint positions for this channel (wave-uniform -> SGPRs) ----
    const int base_j = (c < NJ) ? c : 3 * (c - NJ);
    const int cnt    = (c < NJ) ? 1 : 3;
    int jx[3], jy[3], jv[3];
    int anyvis = 0;
    #pragma unroll
    for (int q = 0; q < 3; ++q) {
        bool use = (q < cnt);
        int  j   = base_j + q;
        float t0 = use ? t[((long long)b * NJ + j) * 2 + 0] : 0.f;
        float t1 = use ? t[((long long)b * NJ + j) * 2 + 1] : 0.f;
        int  vis = use ? v[((long long)b * NJ + j) * 2 + 0] : 0;
        int xi = (int)(t0 * (float)COL); xi = min(max(xi, 0), COL - 1);
        int yi = (int)(t1 * (float)COL); yi = min(max(yi, 0), COL - 1);
        // values are identical across the wave: pin them to SGPRs
        jx[q] = __builtin_amdgcn_readfirstlane(xi);
        jy[q] = __builtin_amdgcn_readfirstlane(yi);
        jv[q] = __builtin_amdgcn_readfirstlane((vis == 1) ? 1 : 0);
        anyvis |= jv[q];
    }
    if (!anyvis) return;                          // masked-out channel: contributes 0 (scalar branch)

    const int n  = lane & 15;                     // N index / A row
    const int kh = (lane >> 4) * 2;               // K sub-offset for second lane half
    const int mh = (lane >> 4) * 8;               // M offset for D rows in second lane half

    // ---- branchless few-hot: per-lane bitmask of rows K with hot[K][n] == 1 ----
    unsigned rowmask = 0u;
    #pragma unroll
    for (int q = 0; q < 3; ++q) {
        unsigned hit = (unsigned)(jv[q] & (jx[q] == n ? 1 : 0));
        rowmask |= (hit << jy[q]);                // hit==0 contributes nothing
    }

    // ---- stage 1: tmp = G(16x16) * hot(16x16), chained 16x16x4 f32 WMMA ----
    v8f tmp = {};
    #pragma unroll
    for (int k0 = 0; k0 < 16; k0 += 4) {
        v2f a, bb;
        a[0] = Gs[n][k0 + kh + 0];                // A: lane row = L&15, K = k0+kh{+1}
        a[1] = Gs[n][k0 + kh + 1];
        bb[0] = (float)((rowmask >> (k0 + kh + 0)) & 1u);
        bb[1] = (float)((rowmask >> (k0 + kh + 1)) & 1u);
        tmp = __builtin_amdgcn_wmma_f32_16x16x4_f32(false, a, false, bb,
                                                    (short)0, tmp, false, false);
    }

    // ---- D-layout -> A-layout relayout through per-wave LDS tile ----
    #pragma unroll
    for (int r = 0; r < 8; ++r)
        Tile[wave][r + mh][n] = tmp[r];           // same-wave LDS ordering handles the dependency

    // ---- stage 2: filt = tmp * G^T ----
    v8f filt = {};
    #pragma unroll
    for (int k0 = 0; k0 < 16; k0 += 4) {
        v2f a, bb;
        a[0]  = Tile[wave][n][k0 + kh + 0];
        a[1]  = Tile[wave][n][k0 + kh + 1];
        bb[0] = Gs[n][k0 + kh + 0];               // (G^T)[K][n] = G[n][K]
        bb[1] = Gs[n][k0 + kh + 1];
        filt = __builtin_amdgcn_wmma_f32_16x16x4_f32(false, a, false, bb,
                                                     (short)0, filt, false, false);
    }

    // ---- per-channel min/max over the valid 14x14 region ----
    float mn =  3.402823466e38f;
    float mx = -3.402823466e38f;
    if (n < COL) {
        #pragma unroll
        for (int r = 0; r < 8; ++r) {
            int M = r + mh;
            if (M < COL) {
                float f = filt[r];
                mn = fminf(mn, f);
                mx = fmaxf(mx, f);
            }
        }
    }
    #pragma unroll
    for (int off = 16; off; off >>= 1) {
        mn = fminf(mn, __shfl_xor(mn, off, 32));
        mx = fmaxf(mx, __shfl_xor(mx, off, 32));
    }
    float denom = mx - mn;
    if (denom == 0.f) denom = 1.f;

    // ---- residual: sum over 14x14 of (h - normalized filt)^2 ----
    const float* hb = h + ((long long)b * CCH + c) * (COL * COL);
    float s = 0.f;
    if (n < COL) {
        #pragma unroll
        for (int r = 0; r < 8; ++r) {
            int M = r + mh;
            if (M < COL) {
                float tt = (filt[r] - mn) / denom;
                float d  = hb[M * COL + n] - tt;
                s += d * d;
            }
        }
    }
    #pragma unroll
    for (int off = 16; off; off >>= 1) s += __shfl_xor(s, off, 32);
    if (lane == 0) atomicAdd(acc, (double)s);
}

__global__ void finalize_kernel(const double* __restrict__ acc,
                                const int* __restrict__ vsum,
                                float* __restrict__ out) {
    if (threadIdx.x == 0 && blockIdx.x == 0) {
        double n1 = 0.5 * (double)(*vsum);
        out[0] = (float)(*acc / n1);
    }
}

extern "C" void kernel_launch(void* const* d_in, const int* in_sizes, int n_in,
                              void* d_out, int out_size, void* d_ws, size_t ws_size,
                              hipStream_t stream) {
    // inputs: d_in[0]=os (UNUSED by the reference), d_in[1]=h, d_in[2]=t, d_in[3]=v
    const float* h = (const float*)d_in[1];
    const float* t = (const float*)d_in[2];
    const int*   v = (const int*)d_in[3];
    float* out = (float*)d_out;

    double* acc  = (double*)d_ws;
    int*    vsum = (int*)((char*)d_ws + 8);

    const int batch = in_sizes[1] / (CCH * COL * COL);
    const int nv    = in_sizes[3];

    init_ws_kernel<<<1, 1, 0, stream>>>(acc, vsum);
    vsum_kernel<<<256, 256, 0, stream>>>(v, nv, vsum);

    const long long total_waves = (long long)batch * CCH;
    const int nblocks = (int)((total_waves + WAVES_PER_BLOCK - 1) / WAVES_PER_BLOCK);
    heatmap_mse_kernel<<<nblocks, 32 * WAVES_PER_BLOCK, 0, stream>>>(h, t, v, acc, batch);

    finalize_kernel<<<1, 1, 0, stream>>>(acc, vsum, out);
}